// Model_27127013441550
// MI455X (gfx1250) — compile-verified
//
#include <hip/hip_runtime.h>
#include <math.h>

// ---------------------------------------------------------------------------
// Point-cloud splat rasterizer for MI455X (gfx1250, wave32).
//
// d2(pixel, point) is computed as a rank-4 GEMM on V_WMMA_F32_16X16X4_F32:
//   A (16 points x 4) rows  = [-2px, -2py, px^2+py^2, 1]
//   B (4 x 16 pixels) cols  = [xf, yf, 1, xf^2+yf^2]
//   D[m][n] = px^2+py^2 + xf^2+yf^2 - 2px*xf - 2py*yf = d2
// D layout (ISA 7.12.2): lane n (0-15) holds rows M=0..7 in its 8 VGPRs,
// lane n+16 holds M=8..15 for the same pixel column n -> each lane gets 8
// candidate d2 values for ITS OWN pixel, perfect for per-lane top-K insertion.
// ---------------------------------------------------------------------------

#define NPTS   4096
#define IMG    128
#define NVIEW  4
#define R2F    (0.02f * 0.02f)
#define FINF   (__builtin_inff())
#define PI_D   3.14159265358979323846

typedef __attribute__((ext_vector_type(2))) float v2f;
typedef __attribute__((ext_vector_type(8))) float v8f;

struct ViewXf {
  float R[NVIEW][9];   // R[v][j*3+i] : view_i = sum_j p_j * R[j][i] + T[i]
  float T[NVIEW][3];
};

// ---- transform points into each view + sigmoid colors ----------------------
__global__ __launch_bounds__(256) void prep_kernel(
    const float* __restrict__ pcd, const float* __restrict__ displace,
    const float* __restrict__ initc,
    float* __restrict__ Px, float* __restrict__ Py, float* __restrict__ Pz,
    float* __restrict__ Col, float* __restrict__ outColors, ViewXf vx)
{
  int n = blockIdx.x * blockDim.x + threadIdx.x;
  if (n >= NPTS) return;
  float p0 = pcd[3*n+0], p1 = pcd[3*n+1], p2 = pcd[3*n+2];
  float c = 1.0f / (1.0f + expf(-(initc[n] + displace[n])));
  Col[n] = c;
  outColors[n] = c;
#pragma unroll
  for (int v = 0; v < NVIEW; ++v) {
    const float* R = vx.R[v];
    const float* T = vx.T[v];
    float x = p0*R[0] + p1*R[3] + p2*R[6] + T[0];
    float y = p0*R[1] + p1*R[4] + p2*R[7] + T[1];
    float z = p0*R[2] + p1*R[5] + p2*R[8] + T[2];
    Px[v*NPTS + n] = x;
    Py[v*NPTS + n] = y;
    Pz[v*NPTS + n] = (z - 0.01f) / (100.0f - 0.01f);
  }
}

// ---- WMMA rasterize + top-5 select + alpha composite -----------------------
__global__ __launch_bounds__(256) void raster_kernel(
    const float* __restrict__ Px, const float* __restrict__ Py,
    const float* __restrict__ Pz, const float* __restrict__ Col,
    float* __restrict__ images)
{
  __shared__ float spx[NPTS];
  __shared__ float spy[NPTS];
  __shared__ float spz[NPTS];
  __shared__ float scol[NPTS];

  const int v         = blockIdx.x >> 7;     // 128 blocks per view
  const int blkInView = blockIdx.x & 127;
  const int tid       = threadIdx.x;

  const float* px = Px + v*NPTS;
  const float* py = Py + v*NPTS;
  const float* pz = Pz + v*NPTS;
  for (int i = tid; i < NPTS; i += 256) {
    spx[i] = px[i]; spy[i] = py[i]; spz[i] = pz[i]; scol[i] = Col[i];
  }
  __syncthreads();

  const int  lane = tid & 31;
  const int  wave = tid >> 5;
  const bool lo   = (lane < 16);
  const int  n16  = lane & 15;

  // this wave's 16 contiguous pixels (one image row segment)
  const int   pixBase = blkInView*128 + wave*16;
  const int   p  = pixBase + n16;
  const int   xi = p & (IMG-1);
  const int   yi = p >> 7;
  const float xf = 1.0f - (2.0f*(float)xi + 1.0f) * (1.0f/(float)IMG);
  const float yf = 1.0f - (2.0f*(float)yi + 1.0f) * (1.0f/(float)IMG);

  // B operand (4x16, loop invariant): v0 = K0(lo)/K2(hi), v1 = K1(lo)/K3(hi)
  v2f bmat;
  bmat.x = lo ? xf : 1.0f;
  bmat.y = lo ? yf : (xf*xf + yf*yf);

  float s0=FINF, s1=FINF, s2=FINF, s3=FINF, s4=FINF;   // top-5 (score = z)
  int   i0=0,    i1=0,    i2=0,    i3=0,    i4=0;
  const int rowoff = lo ? 0 : 8;

  for (int c = 0; c < NPTS/16; ++c) {
    const int base = c*16;
    const float pxv = spx[base + n16];   // LDS broadcast within half-wave
    const float pyv = spy[base + n16];
    // A operand (16x4): v0 = K0(lo)/K2(hi), v1 = K1(lo)/K3(hi)
    v2f amat;
    amat.x = lo ? (-2.0f*pxv) : fmaf(pxv, pxv, pyv*pyv);
    amat.y = lo ? (-2.0f*pyv) : 1.0f;
    v8f acc = {0.f,0.f,0.f,0.f,0.f,0.f,0.f,0.f};
    v8f d = __builtin_amdgcn_wmma_f32_16x16x4_f32(
        /*neg_a=*/false, amat, /*neg_b=*/false, bmat,
        /*c_mod=*/(short)0, acc, /*reuse_a=*/false, /*reuse_b=*/false);

    bool anyhit = false;
#pragma unroll
    for (int r = 0; r < 8; ++r) anyhit = anyhit | (d[r] < R2F);

    if (__ballot(anyhit)) {                       // ~5% of chunks have a hit
#pragma unroll
      for (int r = 0; r < 8; ++r) {
        const int   m  = base + rowoff + r;
        const float z  = spz[m];
        const bool  ok = (d[r] < R2F) && (z > 0.0f);
        const float sc = ok ? z : FINF;
        const bool c0 = sc < s0, c1 = sc < s1, c2 = sc < s2,
                   c3 = sc < s3, c4 = sc < s4;
        s4 = c4 ? (c3 ? s3 : sc) : s4;  i4 = c4 ? (c3 ? i3 : m) : i4;
        s3 = c3 ? (c2 ? s2 : sc) : s3;  i3 = c3 ? (c2 ? i2 : m) : i3;
        s2 = c2 ? (c1 ? s1 : sc) : s2;  i2 = c2 ? (c1 ? i1 : m) : i2;
        s1 = c1 ? (c0 ? s0 : sc) : s1;  i1 = c1 ? (c0 ? i0 : m) : i1;
        s0 = c0 ? sc : s0;              i0 = c0 ? m  : i0;
      }
    }
  }

  // merge with the opposite half-wave (disjoint point subsets, both sorted)
  float b0v = __shfl_xor(s0, 16, 32), b1v = __shfl_xor(s1, 16, 32),
        b2v = __shfl_xor(s2, 16, 32), b3v = __shfl_xor(s3, 16, 32),
        b4v = __shfl_xor(s4, 16, 32);
  int   jb0 = __shfl_xor(i0, 16, 32), jb1 = __shfl_xor(i1, 16, 32),
        jb2 = __shfl_xor(i2, 16, 32), jb3 = __shfl_xor(i3, 16, 32),
        jb4 = __shfl_xor(i4, 16, 32);

  float mv0, mv1, mv2, mv3, mv4;
  int   mi0, mi1, mi2, mi3, mi4;
#define MERGE_ROUND(MV, MI)                                        \
  { const bool ta = (s0 <= b0v);                                   \
    MV = ta ? s0 : b0v;   MI = ta ? i0 : jb0;                      \
    s0  = ta ? s1 : s0;   i0  = ta ? i1 : i0;                      \
    s1  = ta ? s2 : s1;   i1  = ta ? i2 : i1;                      \
    s2  = ta ? s3 : s2;   i2  = ta ? i3 : i2;                      \
    s3  = ta ? s4 : s3;   i3  = ta ? i4 : i3;                      \
    s4  = ta ? FINF : s4;                                          \
    b0v = ta ? b0v : b1v; jb0 = ta ? jb0 : jb1;                    \
    b1v = ta ? b1v : b2v; jb1 = ta ? jb1 : jb2;                    \
    b2v = ta ? b2v : b3v; jb2 = ta ? jb2 : jb3;                    \
    b3v = ta ? b3v : b4v; jb3 = ta ? jb3 : jb4;                    \
    b4v = ta ? b4v : FINF; }
  MERGE_ROUND(mv0, mi0)
  MERGE_ROUND(mv1, mi1)
  MERGE_ROUND(mv2, mi2)
  MERGE_ROUND(mv3, mi3)
  MERGE_ROUND(mv4, mi4)
#undef MERGE_ROUND

  if (lo) {
    float Tacc = 1.0f, pix = 0.0f;
#define COMP(MV, MI)                                               \
    { const bool  val = (MV < FINF);                               \
      const float dx  = spx[MI] - xf;                              \
      const float dy  = spy[MI] - yf;                              \
      const float d2  = dx*dx + dy*dy;  /* exact, matches regather */ \
      const float alpha = val ? (1.0f - d2 / R2F) : 0.0f;          \
      pix  += alpha * Tacc * scol[MI];                             \
      Tacc *= (1.0f - alpha); }
    COMP(mv0, mi0) COMP(mv1, mi1) COMP(mv2, mi2) COMP(mv3, mi3) COMP(mv4, mi4)
#undef COMP
    const int o = ((v*IMG + yi)*IMG + xi)*3;
    images[o+0] = pix; images[o+1] = pix; images[o+2] = pix;
  }
}

// ---------------------------------------------------------------------------
extern "C" void kernel_launch(void* const* d_in, const int* in_sizes, int n_in,
                              void* d_out, int out_size, void* d_ws, size_t ws_size,
                              hipStream_t stream)
{
  const float* pcd      = (const float*)d_in[0];
  const float* displace = (const float*)d_in[1];
  const float* initc    = (const float*)d_in[2];

  float* images    = (float*)d_out;                    // NVIEW*IMG*IMG*3
  float* outColors = images + NVIEW*IMG*IMG*3;         // NPTS

  float* Px  = (float*)d_ws;
  float* Py  = Px + NVIEW*NPTS;
  float* Pz  = Py + NVIEW*NPTS;
  float* Col = Pz + NVIEW*NPTS;                        // total ~208 KB of ws

  // look_at_view_transform(1.5, 15, [0,90,180,270]) in double, like numpy
  ViewXf vx;
  {
    const double dist = 1.5, elev = 15.0 * PI_D / 180.0;
    const double azd[NVIEW] = {0.0, 90.0, 180.0, 270.0};
    const double up[3] = {0.0, 1.0, 0.0};
    for (int v = 0; v < NVIEW; ++v) {
      double az = azd[v] * PI_D / 180.0;
      double C[3] = { dist*cos(elev)*sin(az), dist*sin(elev), dist*cos(elev)*cos(az) };
      double nC = sqrt(C[0]*C[0] + C[1]*C[1] + C[2]*C[2]);
      double z[3] = { -C[0]/nC, -C[1]/nC, -C[2]/nC };
      double x[3] = { up[1]*z[2]-up[2]*z[1], up[2]*z[0]-up[0]*z[2], up[0]*z[1]-up[1]*z[0] };
      double nx = sqrt(x[0]*x[0] + x[1]*x[1] + x[2]*x[2]);
      x[0]/=nx; x[1]/=nx; x[2]/=nx;
      double y[3] = { z[1]*x[2]-z[2]*x[1], z[2]*x[0]-z[0]*x[2], z[0]*x[1]-z[1]*x[0] };
      double Rm[3][3];
      for (int j = 0; j < 3; ++j) { Rm[j][0]=x[j]; Rm[j][1]=y[j]; Rm[j][2]=z[j]; }
      for (int j = 0; j < 3; ++j)
        for (int i = 0; i < 3; ++i)
          vx.R[v][j*3+i] = (float)Rm[j][i];
      for (int i = 0; i < 3; ++i)
        vx.T[v][i] = (float)(-(Rm[0][i]*C[0] + Rm[1][i]*C[1] + Rm[2][i]*C[2]));
    }
  }

  prep_kernel<<<(NPTS + 255)/256, 256, 0, stream>>>(
      pcd, displace, initc, Px, Py, Pz, Col, outColors, vx);
  raster_kernel<<<NVIEW*128, 256, 0, stream>>>(Px, Py, Pz, Col, images);

  (void)in_sizes; (void)n_in; (void)out_size; (void)ws_size;
}